// XELoss_87445534147411
// MI455X (gfx1250) — compile-verified
//
#include <hip/hip_runtime.h>
#include <math.h>

typedef __attribute__((ext_vector_type(2))) float v2f;
typedef __attribute__((ext_vector_type(8))) float v8f;

#define BATCH 8192
#define NWRONG 100
#define NINP 512
#define NSCORE 101            // right + 100 wrong
#define NTILES 7              // ceil(101/16)

// One block per batch row b. 8 waves; waves 0..6 each own a 16-row score tile.
__global__ __launch_bounds__(256) void xeloss_score_kernel(
    const float* __restrict__ feat,
    const float* __restrict__ right,
    const float* __restrict__ wrong,
    float* __restrict__ out)          // out[0]=loss (filled later), out[1+b]=loss_gc[b]
{
    __shared__ float feat_s[NINP];
    __shared__ float scores_s[NTILES * 16];
    __shared__ float red_s[256];

    const int b    = blockIdx.x;
    const int tid  = threadIdx.x;
    const int wave = tid >> 5;
    const int lane = tid & 31;

    // ---- stage feat[b] (2 KB) into LDS, coalesced ----
    for (int i = tid; i < NINP; i += 256)
        feat_s[i] = feat[(size_t)b * NINP + i];
    __syncthreads();

    // ---- WMMA dot products: D = A(16x4 rows of scores) x B(feat bcast) ----
    if (wave < NTILES) {
        const int m = lane & 15;   // row within tile (A: M = lane%16)
        const int h = lane >> 4;   // K-half select (K = vgpr + 2*h)
        const int r = wave * 16 + m;              // global score row
        const float* rowptr;
        if (r == 0)           rowptr = right + (size_t)b * NINP;
        else if (r <= NWRONG) rowptr = wrong + ((size_t)b * NWRONG + (r - 1)) * NINP;
        else                  rowptr = right + (size_t)b * NINP;   // pad rows 101..111 (ignored)

        __builtin_prefetch(rowptr, 0, 0);

        v8f acc = {0.f, 0.f, 0.f, 0.f, 0.f, 0.f, 0.f, 0.f};
        #pragma unroll 4
        for (int k0 = 0; k0 < NINP; k0 += 4) {
            const int kk = k0 + 2 * h;                     // this lane's K pair
            v2f a  = *(const v2f*)(rowptr + kk);           // A[m][kk], A[m][kk+1]
            v2f bb = *(const v2f*)(&feat_s[kk]);           // B[kk][*], B[kk+1][*] (all cols equal)
            acc = __builtin_amdgcn_wmma_f32_16x16x4_f32(
                false, a, false, bb, (short)0, acc, false, false);
        }
        // Column 0 of D: lane 0 -> rows 0..7 (acc[0..7]), lane 16 -> rows 8..15
        if ((lane & 15) == 0) {
            #pragma unroll
            for (int j = 0; j < 8; ++j)
                scores_s[wave * 16 + h * 8 + j] = acc[j];
        }
    }
    __syncthreads();

    // ---- softmax / rank / loss over 101 scores (deterministic LDS trees) ----
    const float sc = (tid < NSCORE) ? scores_s[tid] : -3.402823466e+38f;

    red_s[tid] = sc;
    __syncthreads();
    for (int s = 128; s > 0; s >>= 1) {
        if (tid < s) red_s[tid] = fmaxf(red_s[tid], red_s[tid + s]);
        __syncthreads();
    }
    const float mx = red_s[0];
    __syncthreads();

    const float e = (tid < NSCORE) ? expf(sc - mx) : 0.f;
    red_s[tid] = e;
    __syncthreads();
    for (int s = 128; s > 0; s >>= 1) {
        if (tid < s) red_s[tid] += red_s[tid + s];
        __syncthreads();
    }
    const float sumexp = red_s[0];
    const float s0 = scores_s[0];
    __syncthreads();

    red_s[tid] = (tid < NSCORE && sc > s0) ? 1.f : 0.f;
    __syncthreads();
    for (int s = 128; s > 0; s >>= 1) {
        if (tid < s) red_s[tid] += red_s[tid + s];
        __syncthreads();
    }
    if (tid == 0) {
        const float rank = red_s[0] + 1.f;                 // strictly-greater count + 1
        const float prob = logf(sumexp) - (s0 - mx);       // -log softmax[0]
        const float w    = 2.f - 1.f / rank;
        out[1 + b] = prob * w;
    }
}

// Deterministic single-block mean of out[1..BATCH] -> out[0]
__global__ __launch_bounds__(256) void xeloss_mean_kernel(float* __restrict__ out)
{
    __shared__ float red_s[256];
    const int tid = threadIdx.x;
    float s = 0.f;
    for (int i = tid; i < BATCH; i += 256)
        s += out[1 + i];
    red_s[tid] = s;
    __syncthreads();
    for (int st = 128; st > 0; st >>= 1) {
        if (tid < st) red_s[tid] += red_s[tid + st];
        __syncthreads();
    }
    if (tid == 0) out[0] = red_s[0] / (float)BATCH;
}

extern "C" void kernel_launch(void* const* d_in, const int* in_sizes, int n_in,
                              void* d_out, int out_size, void* d_ws, size_t ws_size,
                              hipStream_t stream)
{
    const float* feat  = (const float*)d_in[0];
    const float* right = (const float*)d_in[1];
    const float* wrong = (const float*)d_in[2];
    float* out = (float*)d_out;

    xeloss_score_kernel<<<BATCH, 256, 0, stream>>>(feat, right, wrong, out);
    xeloss_mean_kernel<<<1, 256, 0, stream>>>(out);
}